// SpectralConv2d_326417514511
// MI455X (gfx1250) — compile-verified
//
#include <hip/hip_runtime.h>

// Problem constants (match reference)
#define Bdim 2
#define Ldim 8
#define Cdim 64
#define Hdim 64
#define Wdim 33
#define M1c  16
#define M2c  16
#define BL   16              // B*L == 16 == WMMA M
#define HW   (Hdim * Wdim)   // 2112

typedef float v2f __attribute__((ext_vector_type(2)));
typedef float v8f __attribute__((ext_vector_type(8)));

// ---------------------------------------------------------------------------
// Kernel 1: vectorized zero fill of the whole output (bandwidth-bound part).
// ---------------------------------------------------------------------------
__global__ __launch_bounds__(256) void zero_fill_f4(float4* __restrict__ out, long n4) {
    long i = (long)blockIdx.x * blockDim.x + threadIdx.x;
    if (i < n4) {
        float4 z; z.x = 0.f; z.y = 0.f; z.z = 0.f; z.w = 0.f;
        out[i] = z;
    }
}

// ---------------------------------------------------------------------------
// Kernel 2: per-pixel complex GEMM on the two frequency corners via
// V_WMMA_F32_16X16X4_F32 (fp32-exact).
//   grid = 2 corners * 256 pixels = 512 blocks, 128 threads (4 waves).
//   wave nt handles output channels o = nt*16 .. nt*16+15.
// ---------------------------------------------------------------------------
__global__ __launch_bounds__(128) void spectral_corner_wmma(
    const float* __restrict__ xr, const float* __restrict__ xi,
    const float* __restrict__ wr1, const float* __restrict__ wi1,
    const float* __restrict__ wr2, const float* __restrict__ wi2,
    float* __restrict__ out)
{
    // +1 padding -> LDS bank-conflict-free row reads
    __shared__ float sXr[BL][Cdim + 1];
    __shared__ float sXi[BL][Cdim + 1];

    const int blk    = blockIdx.x;
    const int corner = blk >> 8;          // 0: low freqs, 1: high freqs (H)
    const int pix    = blk & 255;
    const int px     = pix >> 4;          // weight row index 0..15
    const int py     = pix & 15;          // weight col index 0..15
    const int h      = corner ? (Hdim - M1c + px) : px;

    const int tid = threadIdx.x;

    // Cooperative stage of the X tile (16 bl x 64 c, real+imag) into LDS.
    {
        const long sp = (long)h * Wdim + py;
        for (int i = tid; i < BL * Cdim; i += 128) {
            const int m = i >> 6;         // bl = b*L + l
            const int c = i & 63;
            const long off = ((long)m * Cdim + c) * HW + sp;
            sXr[m][c] = xr[off];
            sXi[m][c] = xi[off];
        }
    }
    __syncthreads();

    const float* __restrict__ wr = corner ? wr2 : wr1;
    const float* __restrict__ wi = corner ? wi2 : wi1;

    const int wave = tid >> 5;            // output-channel tile 0..3
    const int lane = tid & 31;
    const int nloc = lane & 15;           // A: row m; B: col n; D: col n
    const int kb   = (lane >> 4) << 1;    // K sub-pair: 0 (lanes 0-15) / 2 (lanes 16-31)
    const int m    = nloc;
    const int o    = (wave << 4) + nloc;

    // w[o][c][px][py]: c-stride = 256 floats, o-stride = 64*256 floats
    const long wpix = (long)px * M2c + py;
    const float* __restrict__ wrp = wr + (long)o * Cdim * (M1c * M2c) + wpix;
    const float* __restrict__ wip = wi + (long)o * Cdim * (M1c * M2c) + wpix;

    v8f accr = {};   // xr*wr - xi*wi
    v8f acci = {};   // xr*wi + xi*wr

#pragma unroll
    for (int c0 = 0; c0 < Cdim; c0 += 4) {
        const int k0 = c0 + kb;

        // A operands (16x4, M x K) from LDS
        v2f ar, ai, ain;
        ar.x  = sXr[m][k0];     ar.y  = sXr[m][k0 + 1];
        ai.x  = sXi[m][k0];     ai.y  = sXi[m][k0 + 1];
        ain.x = -ai.x;          ain.y = -ai.y;   // f32 WMMA has no A-neg modifier

        // B operands (4x16, K x N): lane holds column o, K pair by lane group
        v2f br, bi;
        br.x = wrp[(long)k0 * 256];       br.y = wrp[(long)(k0 + 1) * 256];
        bi.x = wip[(long)k0 * 256];       bi.y = wip[(long)(k0 + 1) * 256];

        accr = __builtin_amdgcn_wmma_f32_16x16x4_f32(false, ar,  false, br,
                                                     (short)0, accr, false, false);
        accr = __builtin_amdgcn_wmma_f32_16x16x4_f32(false, ain, false, bi,
                                                     (short)0, accr, false, false);
        acci = __builtin_amdgcn_wmma_f32_16x16x4_f32(false, ar,  false, bi,
                                                     (short)0, acci, false, false);
        acci = __builtin_amdgcn_wmma_f32_16x16x4_f32(false, ai,  false, br,
                                                     (short)0, acci, false, false);
    }

    // D layout: lane<16 -> rows 0..7, lane>=16 -> rows 8..15; col = nloc.
    // Output shape (2, B, L, C, H, W), real plane first.
    const long plane = (long)Bdim * Ldim * Cdim * HW;
    const int  rbase = (lane >> 4) << 3;
    const long spo   = (long)h * Wdim + py;
#pragma unroll
    for (int r = 0; r < 8; ++r) {
        const int  bl  = rbase + r;
        const long off = ((long)bl * Cdim + o) * HW + spo;
        out[off]         = accr[r];
        out[plane + off] = acci[r];
    }
}

// ---------------------------------------------------------------------------
extern "C" void kernel_launch(void* const* d_in, const int* in_sizes, int n_in,
                              void* d_out, int out_size, void* d_ws, size_t ws_size,
                              hipStream_t stream) {
    const float* xr  = (const float*)d_in[0];
    const float* xi  = (const float*)d_in[1];
    const float* wr1 = (const float*)d_in[2];
    const float* wi1 = (const float*)d_in[3];
    const float* wr2 = (const float*)d_in[4];
    const float* wi2 = (const float*)d_in[5];
    float* out = (float*)d_out;

    // 1) zero the whole output (out_size = 2*B*L*C*H*W, divisible by 4)
    const long n4 = (long)out_size / 4;
    const int  zgrid = (int)((n4 + 255) / 256);
    zero_fill_f4<<<zgrid, 256, 0, stream>>>((float4*)out, n4);

    // 2) corner compute: 2 corners * 256 pixels, 4 waves per block
    spectral_corner_wmma<<<512, 128, 0, stream>>>(xr, xi, wr1, wi1, wr2, wi2, out);
}